// Statis_layer_23158463660279
// MI455X (gfx1250) — compile-verified
//
#include <hip/hip_runtime.h>
#include <hip/hip_bf16.h>
#include <math.h>

// ---------------- problem constants ----------------
#define BATCH   256
#define SEQ     65536
#define NPATCH  2047          // (65536-64)/32 + 1
#define KDIM    2046          // NPATCH-1
#define KPAD    2048          // padded K for WMMA (multiple of 4)
#define L1      2047
#define L1P     1023
#define L2      1023
#define L2P     511
#define L3      511
#define EPS     1e-5f

typedef float v2f __attribute__((ext_vector_type(2)));
typedef float v8f __attribute__((ext_vector_type(8)));

// ---------------- workspace layout (float offsets) ----------------
// [0,512): BN accumulators: s1(32) q1(32) s2(64) q2(64) s3(128) q3(128)
#define OFF_S1   0
#define OFF_Q1   32
#define OFF_S2   64
#define OFF_Q2   128
#define OFF_S3   192
#define OFF_Q3   320
#define OFF_MMAT 512u                       // (KPAD/2) x BATCH x 2 = 524288
#define OFF_X3   (OFF_MMAT + 524288u)       // B x 3 x NPATCH       = 1572096
#define OFF_Y    2097152u                   // reusable y buffer    (<=16769024)
#define OFF_H    18874368u                  // reusable h buffer    (<=8380416)

// ---------------- kernel 1: patch stats + dmax (K-pair-interleaved) -------
// Mpair layout: element (k, b) stored at (k>>1)*(2*BATCH) + 2*b + (k&1)
// so that (B[k,n], B[k+1,n]) for even k is one contiguous float2.
__global__ __launch_bounds__(256) void k_stats(const float* __restrict__ x,
                                               float* __restrict__ x3,
                                               float* __restrict__ Mpair) {
  int idx = blockIdx.x * 256 + threadIdx.x;          // BATCH * 2048 threads
  if (idx >= BATCH * 2048) return;
  int b = idx >> 11;
  int i = idx & 2047;
  const float4* xp = reinterpret_cast<const float4*>(x + (size_t)b * SEQ + (size_t)i * 32);

  if (i < NPATCH) {
    float s = 0.f, q = 0.f;
#pragma unroll
    for (int j = 0; j < 16; ++j) {
      float4 v = xp[j];
      s += (v.x + v.y) + (v.z + v.w);
      q += (v.x * v.x + v.y * v.y) + (v.z * v.z + v.w * v.w);
    }
    float mean = s * (1.0f / 64.0f);
    float var  = (q - 64.0f * mean * mean) * (1.0f / 63.0f);
    float sd   = sqrtf(fmaxf(var, 0.0f));
    x3[(size_t)b * (3 * NPATCH) + i]          = mean;   // channel 0
    x3[(size_t)b * (3 * NPATCH) + NPATCH + i] = sd;     // channel 1
  }

  float dmax = 0.0f;                         // rows >= KDIM are zero padding
  if (i < KDIM) {
    dmax = -3.402823466e+38f;
#pragma unroll
    for (int j = 0; j < 16; ++j) {
      float4 v0 = xp[j];
      float4 v1 = xp[j + 8];                 // +32 floats
      dmax = fmaxf(dmax, fmaxf(fmaxf(v1.x - v0.x, v1.y - v0.y),
                               fmaxf(v1.z - v0.z, v1.w - v0.w)));
    }
  }
  Mpair[(size_t)(i >> 1) * (2 * BATCH) + 2 * b + (i & 1)] = dmax;
}

// ---------------- kernel 2: WMMA fp32 GEMM  D[o,b] = W[o,:]·M[:,b] + bias ----
// one wave per block; wave computes 16(M) x 64(N) with 4 accumulators.
// Branch-free inner loop: clamped addresses, all loads are b64.
__global__ __launch_bounds__(32) void k_gemm(const float* __restrict__ W,
                                             const float* __restrict__ bias,
                                             const float* __restrict__ Mpair,
                                             float* __restrict__ x3) {
  const int lane = threadIdx.x;              // 0..31
  const int m0 = blockIdx.x * 16;            // 128 M-tiles
  const int n0 = blockIdx.y * 64;            // 4 N-groups
  const int mr = lane & 15;
  const int hi = lane >> 4;                  // 0/1 -> K half
  int mrow = m0 + mr;
  if (mrow > NPATCH - 1) mrow = NPATCH - 1;  // clamp: garbage row never stored
  const size_t wbase = (size_t)mrow * KDIM;  // even
  const float2* Bp = reinterpret_cast<const float2*>(Mpair);  // [KPAD/2][BATCH]
  const int nb = n0 + mr;

  v8f c0 = {}, c1 = {}, c2 = {}, c3 = {};
  for (int k0 = 0; k0 < KPAD; k0 += 4) {
    int k = k0 + (hi << 1);                  // even
    int kc = (k < KDIM) ? k : 0;             // clamp: B rows >= KDIM are zero
    float2 af = *reinterpret_cast<const float2*>(W + wbase + kc);  // b64
    v2f a; a.x = af.x; a.y = af.y;
    const float2* Bk = Bp + (size_t)(k >> 1) * BATCH + nb;
    float2 f0 = Bk[0];
    float2 f1 = Bk[16];
    float2 f2 = Bk[32];
    float2 f3 = Bk[48];
    v2f b0; b0.x = f0.x; b0.y = f0.y;
    v2f b1; b1.x = f1.x; b1.y = f1.y;
    v2f b2; b2.x = f2.x; b2.y = f2.y;
    v2f b3; b3.x = f3.x; b3.y = f3.y;
    c0 = __builtin_amdgcn_wmma_f32_16x16x4_f32(false, a, false, b0, (short)0, c0, false, false);
    c1 = __builtin_amdgcn_wmma_f32_16x16x4_f32(false, a, false, b1, (short)0, c1, false, false);
    c2 = __builtin_amdgcn_wmma_f32_16x16x4_f32(false, a, false, b2, (short)0, c2, false, false);
    c3 = __builtin_amdgcn_wmma_f32_16x16x4_f32(false, a, false, b3, (short)0, c3, false, false);
  }

  // D layout: VGPR r -> M = r (lanes 0-15) / r+8 (lanes 16-31); N = lane&15
  const int mhi = hi * 8;
#pragma unroll
  for (int r = 0; r < 8; ++r) {
    int m = m0 + r + mhi;
    if (m < NPATCH) {
      float bv = bias[m];
      size_t o;
      o = (size_t)(n0 +  0 + mr) * (3 * NPATCH) + 2 * NPATCH + m; x3[o] = c0[r] + bv;
      o = (size_t)(n0 + 16 + mr) * (3 * NPATCH) + 2 * NPATCH + m; x3[o] = c1[r] + bv;
      o = (size_t)(n0 + 32 + mr) * (3 * NPATCH) + 2 * NPATCH + m; x3[o] = c2[r] + bv;
      o = (size_t)(n0 + 48 + mr) * (3 * NPATCH) + 2 * NPATCH + m; x3[o] = c3[r] + bv;
    }
  }
}

// ---------------- block reduction helper (blockDim == 256) ----------------
__device__ __forceinline__ float block_reduce_sum(float v, float* red) {
  int t = threadIdx.x;
  red[t] = v; __syncthreads();
  for (int off = 128; off > 0; off >>= 1) {
    if (t < off) red[t] += red[t + off];
    __syncthreads();
  }
  float r = red[0];
  __syncthreads();
  return r;
}

// ---------------- kernel 3: conv1 (3->32, k=5, pad=2) + BN partial sums ----
__global__ __launch_bounds__(256) void k_conv1(const float* __restrict__ x3,
                                               const float* __restrict__ w,
                                               const float* __restrict__ bias,
                                               float* __restrict__ y,
                                               float* __restrict__ s_acc,
                                               float* __restrict__ q_acc) {
  int b = blockIdx.x;       // 256
  int c = blockIdx.y;       // 32
  __shared__ float wsh[16];
  __shared__ float red[256];
  if (threadIdx.x < 15) wsh[threadIdx.x] = w[c * 15 + threadIdx.x];
  __syncthreads();
  const float* xin = x3 + (size_t)b * (3 * L1);
  float bv = bias[c];
  float ls = 0.f, lq = 0.f;
  for (int l = threadIdx.x; l < L1; l += 256) {
    float acc = bv;
#pragma unroll
    for (int ic = 0; ic < 3; ++ic) {
      const float* row = xin + ic * L1;
      const float* wr = wsh + ic * 5;
#pragma unroll
      for (int t = 0; t < 5; ++t) {
        int p = l + t - 2;
        float v = (p >= 0 && p < L1) ? row[p] : 0.0f;
        acc = fmaf(wr[t], v, acc);
      }
    }
    y[((size_t)b * 32 + c) * L1 + l] = acc;
    ls += acc;
    lq = fmaf(acc, acc, lq);
  }
  float rs = block_reduce_sum(ls, red);
  float rq = block_reduce_sum(lq, red);
  if (threadIdx.x == 0) { atomicAdd(&s_acc[c], rs); atomicAdd(&q_acc[c], rq); }
}

// ---------------- generic 3-tap conv (pad=1) + BN partial sums -------------
__global__ __launch_bounds__(256) void k_conv3tap(const float* __restrict__ hin,
                                                  const float* __restrict__ w,
                                                  const float* __restrict__ bias,
                                                  float* __restrict__ y,
                                                  float* __restrict__ s_acc,
                                                  float* __restrict__ q_acc,
                                                  int Cin, int L) {
  int b = blockIdx.x;
  int c = blockIdx.y;       // gridDim.y == Cout
  __shared__ float wsh[192];        // up to 64*3
  __shared__ float red[256];
  for (int j = threadIdx.x; j < Cin * 3; j += 256) wsh[j] = w[(size_t)c * Cin * 3 + j];
  __syncthreads();
  const float* xin = hin + (size_t)b * Cin * L;
  float bv = bias[c];
  float ls = 0.f, lq = 0.f;
  for (int l = threadIdx.x; l < L; l += 256) {
    float acc = bv;
    for (int ic = 0; ic < Cin; ++ic) {
      const float* row = xin + (size_t)ic * L;
      float vm = (l > 0)     ? row[l - 1] : 0.0f;
      float v0 = row[l];
      float vp = (l < L - 1) ? row[l + 1] : 0.0f;
      acc = fmaf(wsh[ic * 3 + 0], vm, acc);
      acc = fmaf(wsh[ic * 3 + 1], v0, acc);
      acc = fmaf(wsh[ic * 3 + 2], vp, acc);
    }
    y[((size_t)b * gridDim.y + c) * L + l] = acc;
    ls += acc;
    lq = fmaf(acc, acc, lq);
  }
  float rs = block_reduce_sum(ls, red);
  float rq = block_reduce_sum(lq, red);
  if (threadIdx.x == 0) { atomicAdd(&s_acc[c], rs); atomicAdd(&q_acc[c], rq); }
}

// ---------------- BN (batch stats) + relu + maxpool2 ----------------------
__global__ __launch_bounds__(256) void k_bnpool(const float* __restrict__ y,
                                                const float* __restrict__ s_acc,
                                                const float* __restrict__ q_acc,
                                                const float* __restrict__ g,
                                                const float* __restrict__ be,
                                                float* __restrict__ h,
                                                int C, int L, int Lp, float invCount) {
  int idx = blockIdx.x * 256 + threadIdx.x;      // BATCH*C*Lp
  if (idx >= BATCH * C * Lp) return;
  int lp = idx % Lp;
  int c  = (idx / Lp) % C;
  int b  = idx / (Lp * C);
  float m = s_acc[c] * invCount;
  float v = q_acc[c] * invCount - m * m;
  float scale = g[c] * rsqrtf(v + EPS);
  float shift = be[c] - m * scale;
  const float* yr = y + ((size_t)b * C + c) * L + 2 * lp;
  float z0 = fmaxf(fmaf(yr[0], scale, shift), 0.0f);
  float z1 = fmaxf(fmaf(yr[1], scale, shift), 0.0f);
  h[((size_t)b * C + c) * Lp + lp] = fmaxf(z0, z1);
}

// ---------------- final: BN + relu + mean over length ---------------------
__global__ __launch_bounds__(256) void k_out(const float* __restrict__ y3,
                                             const float* __restrict__ s_acc,
                                             const float* __restrict__ q_acc,
                                             const float* __restrict__ g,
                                             const float* __restrict__ be,
                                             float* __restrict__ out) {
  int b = blockIdx.x;       // 256
  int c = blockIdx.y;       // 128
  const float invCount = 1.0f / (256.0f * 511.0f);
  float m = s_acc[c] * invCount;
  float v = q_acc[c] * invCount - m * m;
  float scale = g[c] * rsqrtf(v + EPS);
  float shift = be[c] - m * scale;
  const float* yr = y3 + ((size_t)b * 128 + c) * L3;
  float ls = 0.f;
  for (int l = threadIdx.x; l < L3; l += 256)
    ls += fmaxf(fmaf(yr[l], scale, shift), 0.0f);
  __shared__ float red[256];
  float rs = block_reduce_sum(ls, red);
  if (threadIdx.x == 0) out[(size_t)b * 128 + c] = rs * (1.0f / 511.0f);
}

// ---------------- launcher ----------------
extern "C" void kernel_launch(void* const* d_in, const int* in_sizes, int n_in,
                              void* d_out, int out_size, void* d_ws, size_t ws_size,
                              hipStream_t stream) {
  const float* x       = (const float*)d_in[0];
  const float* conv1_w = (const float*)d_in[1];
  const float* conv1_b = (const float*)d_in[2];
  const float* w1      = (const float*)d_in[3];
  const float* b1      = (const float*)d_in[4];
  const float* g1      = (const float*)d_in[5];
  const float* be1     = (const float*)d_in[6];
  const float* w2      = (const float*)d_in[7];
  const float* b2      = (const float*)d_in[8];
  const float* g2      = (const float*)d_in[9];
  const float* be2     = (const float*)d_in[10];
  const float* w3      = (const float*)d_in[11];
  const float* b3      = (const float*)d_in[12];
  const float* g3      = (const float*)d_in[13];
  const float* be3     = (const float*)d_in[14];
  float* out = (float*)d_out;

  float* ws    = (float*)d_ws;
  float* Mpair = ws + OFF_MMAT;
  float* x3    = ws + OFF_X3;
  float* y     = ws + OFF_Y;
  float* h     = ws + OFF_H;
  float* s1 = ws + OFF_S1; float* q1 = ws + OFF_Q1;
  float* s2 = ws + OFF_S2; float* q2 = ws + OFF_Q2;
  float* s3 = ws + OFF_S3; float* q3 = ws + OFF_Q3;

  // zero BN accumulators (graph-capturable)
  hipMemsetAsync(d_ws, 0, 512 * sizeof(float), stream);

  // 1) patch stats + dmax
  k_stats<<<(BATCH * 2048 + 255) / 256, 256, 0, stream>>>(x, x3, Mpair);

  // 2) WMMA fp32 GEMM -> x3 channel 2
  k_gemm<<<dim3(128, 4), 32, 0, stream>>>(conv1_w, conv1_b, Mpair, x3);

  // 3) conv1 + BN1 stats
  k_conv1<<<dim3(BATCH, 32), 256, 0, stream>>>(x3, w1, b1, y, s1, q1);
  // 4) bn1 + relu + pool -> h1 (B,32,1023)
  k_bnpool<<<(BATCH * 32 * L1P + 255) / 256, 256, 0, stream>>>(
      y, s1, q1, g1, be1, h, 32, L1, L1P, 1.0f / (256.0f * 2047.0f));

  // 5) conv2 + BN2 stats
  k_conv3tap<<<dim3(BATCH, 64), 256, 0, stream>>>(h, w2, b2, y, s2, q2, 32, L2);
  // 6) bn2 + relu + pool -> h2 (B,64,511)
  k_bnpool<<<(BATCH * 64 * L2P + 255) / 256, 256, 0, stream>>>(
      y, s2, q2, g2, be2, h, 64, L2, L2P, 1.0f / (256.0f * 1023.0f));

  // 7) conv3 + BN3 stats
  k_conv3tap<<<dim3(BATCH, 128), 256, 0, stream>>>(h, w3, b3, y, s3, q3, 64, L3);
  // 8) bn3 + relu + mean -> out (B,128)
  k_out<<<dim3(BATCH, 128), 256, 0, stream>>>(y, s3, q3, g3, be3, out);

  (void)in_sizes; (void)n_in; (void)out_size; (void)ws_size;
}